// DownSample_5446018532080
// MI455X (gfx1250) — compile-verified
//
#include <hip/hip_runtime.h>

#define BATCH  2
#define LSEQ   2048
#define DMODEL 128
#define DINNER 256
#define NST    16
#define DTRANK 8
#define ROWS   (BATCH*LSEQ)          /* 4096 */
#define XPN    (DTRANK + 2*NST)      /* 40   */

typedef __attribute__((ext_vector_type(2))) float v2f;
typedef __attribute__((ext_vector_type(8))) float v8f;

__device__ __forceinline__ float silu_f(float x) { return x / (1.f + __expf(-x)); }

// ---------------- downsampling conv (K=4, stride=4) + SiLU + NCH->NLC ----------------
__global__ void k_downsample(const float* __restrict__ x, const float* __restrict__ w,
                             const float* __restrict__ bias, float* __restrict__ h)
{
    int t = blockIdx.x * 256 + threadIdx.x;
    if (t >= BATCH * LSEQ * DMODEL) return;
    int o = t & (DMODEL - 1);
    int l = (t >> 7) & (LSEQ - 1);
    int b = t >> 18;
    const float* xb = x + (size_t)b * 64 * 8192 + l * 4;
    const float* wo = w + o * 64 * 4;
    float acc = bias[o];
#pragma unroll 4
    for (int ci = 0; ci < 64; ++ci) {
        const float* xr = xb + ci * 8192;
        const float* wr = wo + ci * 4;
        acc += xr[0]*wr[0] + xr[1]*wr[1] + xr[2]*wr[2] + xr[3]*wr[3];
    }
    h[t] = silu_f(acc);
}

// ---------------- reverse along L ----------------
__global__ void k_reverse(const float* __restrict__ src, float* __restrict__ dst)
{
    int t = blockIdx.x * 256 + threadIdx.x;
    if (t >= BATCH * LSEQ * DMODEL) return;
    int d = t & (DMODEL - 1);
    int l = (t >> 7) & (LSEQ - 1);
    int b = t >> 18;
    dst[t] = src[((size_t)b * LSEQ + (LSEQ - 1 - l)) * DMODEL + d];
}

// ---------------- residual add + LayerNorm (one wave per 128-elem row) ----------------
__global__ void k_resid_ln(const float* __restrict__ cur, float* __restrict__ res,
                           float* __restrict__ hn, const float* __restrict__ w,
                           const float* __restrict__ bb, int first)
{
    int wid  = (blockIdx.x * 256 + threadIdx.x) >> 5;
    int lane = threadIdx.x & 31;
    if (wid >= ROWS) return;
    const float* c = cur + (size_t)wid * DMODEL;
    float* rp = res + (size_t)wid * DMODEL;
    float v[4]; float s = 0.f;
#pragma unroll
    for (int i = 0; i < 4; ++i) {
        int idx = lane + 32 * i;
        float t = c[idx] + (first ? 0.f : rp[idx]);
        v[i] = t; rp[idx] = t; s += t;
    }
#pragma unroll
    for (int off = 16; off; off >>= 1) s += __shfl_xor(s, off, 32);
    float mean = s * (1.f / DMODEL);
    float vs = 0.f;
#pragma unroll
    for (int i = 0; i < 4; ++i) { float d = v[i] - mean; vs += d * d; }
#pragma unroll
    for (int off = 16; off; off >>= 1) vs += __shfl_xor(vs, off, 32);
    float rstd = rsqrtf(vs * (1.f / DMODEL) + 1e-5f);
#pragma unroll
    for (int i = 0; i < 4; ++i) {
        int idx = lane + 32 * i;
        hn[(size_t)wid * DMODEL + idx] = (v[i] - mean) * rstd * w[idx] + bb[idx];
    }
}

// ---------------- generic fp32 WMMA GEMM: C(ROWS x N) = A(ROWS x K) * W(N x K)^T -----
// one wave owns one 16x16 tile, V_WMMA_F32_16X16X4_F32 over K in steps of 4
__global__ void k_gemm_wmma(const float* __restrict__ A, int lda,
                            const float* __restrict__ W, int ldw, int N,
                            float* __restrict__ C, int ldc,
                            int K, int tilesN, int numTiles)
{
    int wave = (blockIdx.x * 256 + threadIdx.x) >> 5;
    if (wave >= numTiles) return;                 // wave-uniform: EXEC stays all-ones
    int lane = threadIdx.x & 31;
    int g = lane >> 4, r = lane & 15;
    int mt = wave / tilesN, nt = wave % tilesN;
    int col = nt * 16 + r;
    int colOK = col < N;
    const float* Arow = A + (size_t)(mt * 16 + r) * lda + 2 * g;
    const float* Wrow = W + (size_t)(colOK ? col : 0) * ldw + 2 * g;
    v8f acc = {};
    for (int k0 = 0; k0 < K; k0 += 4) {
        v2f av = *(const v2f*)(Arow + k0);
        v2f bv = *(const v2f*)(Wrow + k0);
        if (!colOK) { bv.x = 0.f; bv.y = 0.f; }   // lane-level select, not EXEC
        acc = __builtin_amdgcn_wmma_f32_16x16x4_f32(false, av, false, bv,
                                                    (short)0, acc, false, false);
    }
    if (colOK) {
        float* Cp = C + (size_t)(mt * 16 + 8 * g) * ldc + col;
#pragma unroll
        for (int i = 0; i < 8; ++i) Cp[(size_t)i * ldc] = acc[i];
    }
}

// ---------------- causal depthwise conv (width 4) + bias + SiLU ----------------
__global__ void k_dwconv_silu(const float* __restrict__ xz, const float* __restrict__ cw,
                              const float* __restrict__ cb, float* __restrict__ xc)
{
    int t = blockIdx.x * 256 + threadIdx.x;
    if (t >= ROWS * DINNER) return;
    int e  = t & (DINNER - 1);
    int bl = t >> 8;
    int l  = bl & (LSEQ - 1);
    float acc = cb[e];
    const float* w4 = cw + e * 4;
#pragma unroll
    for (int k = 0; k < 4; ++k) {
        int ls = l - 3 + k;
        if (ls >= 0) acc += xz[(size_t)(bl - 3 + k) * (2 * DINNER) + e] * w4[k];
    }
    xc[t] = silu_f(acc);
}

// ---------------- dt = softplus(dtproj_out + dt_bias) ----------------
__global__ void k_softplus_bias(float* __restrict__ dtf, const float* __restrict__ dtb)
{
    int t = blockIdx.x * 256 + threadIdx.x;
    if (t >= ROWS * DINNER) return;
    float x = dtf[t] + dtb[t & (DINNER - 1)];
    dtf[t] = (x > 20.f) ? x : log1pf(__expf(x));
}

// ---------------- selective scan: lane = (b,d,n); 16-lane reduce for y ----------------
__global__ void k_ssm_scan(const float* __restrict__ dtf, const float* __restrict__ xc,
                           const float* __restrict__ dbc, const float* __restrict__ xz,
                           const float* __restrict__ Alog, const float* __restrict__ Dp,
                           float* __restrict__ ybuf)
{
    int t = blockIdx.x * 256 + threadIdx.x;     // 8192 lanes total
    int n = t & (NST - 1);
    int d = (t >> 4) & (DINNER - 1);
    int b = t >> 12;
    float a_dn = -__expf(Alog[d * NST + n]);
    float Dpd  = Dp[d];
    float hst  = 0.f;
    for (int l = 0; l < LSEQ; ++l) {
        int bl = b * LSEQ + l;
        float dt = dtf[(size_t)bl * DINNER + d];
        float xv = xc [(size_t)bl * DINNER + d];
        const float* row = dbc + (size_t)bl * XPN;
        float Bn = row[DTRANK + n];
        float Cn = row[DTRANK + NST + n];
        float dA = __expf(dt * a_dn);
        hst = fmaf(dA, hst, dt * Bn * xv);
        float yv = hst * Cn;
        yv += __shfl_xor(yv, 8, 16);
        yv += __shfl_xor(yv, 4, 16);
        yv += __shfl_xor(yv, 2, 16);
        yv += __shfl_xor(yv, 1, 16);
        if (n == 0) {
            float zv = xz[(size_t)bl * (2 * DINNER) + DINNER + d];
            ybuf[(size_t)bl * DINNER + d] = (yv + Dpd * xv) * silu_f(zv);
        }
    }
}

// ---------------- stack epilogues ----------------
__global__ void k_finish_fwd(const float* __restrict__ cur, const float* __restrict__ res,
                             float* __restrict__ sumf, float* __restrict__ rf_out)
{
    int t = blockIdx.x * 256 + threadIdx.x;
    if (t >= BATCH * LSEQ * DMODEL) return;
    float rv = res[t];
    rf_out[t] = rv;
    sumf[t] = cur[t] + rv;
}

__global__ void k_finish_bwd(const float* __restrict__ cur, const float* __restrict__ res,
                             const float* __restrict__ sumf, float* __restrict__ out0,
                             float* __restrict__ rb_out)
{
    int t = blockIdx.x * 256 + threadIdx.x;
    if (t >= BATCH * LSEQ * DMODEL) return;
    int d = t & (DMODEL - 1);
    int l = (t >> 7) & (LSEQ - 1);
    int b = t >> 18;
    float rv = res[t];
    rb_out[t] = rv;                                   // rb kept in backward-time order
    int tr = (b * LSEQ + (LSEQ - 1 - l)) * DMODEL + d;
    out0[((size_t)b * DMODEL + d) * LSEQ + l] = sumf[t] + (cur[tr] + res[tr]);
}

// ---------------- host side ----------------
static inline void launch_gemm(const float* A, int lda, const float* W, int ldw, int N,
                               float* C, int ldc, int K, hipStream_t s)
{
    int tilesN = (N + 15) / 16;
    int numTiles = (ROWS / 16) * tilesN;
    int blocks = (numTiles + 7) / 8;                  // 8 waves (tiles) per 256-thread block
    k_gemm_wmma<<<blocks, 256, 0, s>>>(A, lda, W, ldw, N, C, ldc, K, tilesN, numTiles);
}

extern "C" void kernel_launch(void* const* d_in, const int* in_sizes, int n_in,
                              void* d_out, int out_size, void* d_ws, size_t ws_size,
                              hipStream_t stream)
{
    const float* x         = (const float*)d_in[0];
    const float* convd_w   = (const float*)d_in[1];
    const float* convd_b   = (const float*)d_in[2];
    const float* ln_w      = (const float*)d_in[3];
    const float* ln_b      = (const float*)d_in[4];
    const float* in_proj_w = (const float*)d_in[5];
    const float* conv_w    = (const float*)d_in[6];
    const float* conv_b    = (const float*)d_in[7];
    const float* xproj_w   = (const float*)d_in[8];
    const float* dtproj_w  = (const float*)d_in[9];
    const float* dtproj_b  = (const float*)d_in[10];
    const float* A_log     = (const float*)d_in[11];
    const float* Dparam    = (const float*)d_in[12];
    const float* outproj_w = (const float*)d_in[13];

    const size_t NH = (size_t)BATCH * LSEQ * DMODEL;  // 524288
    float* out0 = (float*)d_out;
    float* rf_o = out0 + NH;
    float* rb_o = rf_o + NH;

    float* ws   = (float*)d_ws;
    float* h_   = ws;
    float* hrev = h_   + NH;
    float* cur  = hrev + NH;
    float* res  = cur  + NH;
    float* hn   = res  + NH;
    float* sumf = hn   + NH;
    float* xz   = sumf + NH;                              // ROWS x 512
    float* xcb  = xz   + (size_t)ROWS * 2 * DINNER;       // ROWS x 256
    float* dbc  = xcb  + (size_t)ROWS * DINNER;           // ROWS x 40
    float* dtf  = dbc  + (size_t)ROWS * XPN;              // ROWS x 256
    float* ybuf = dtf  + (size_t)ROWS * DINNER;           // ROWS x 256

    k_downsample<<<NH / 256, 256, 0, stream>>>(x, convd_w, convd_b, h_);
    k_reverse<<<NH / 256, 256, 0, stream>>>(h_, hrev);

    for (int dir = 0; dir < 2; ++dir) {
        const float* input = dir ? hrev : h_;
        for (int i = 0; i < 3; ++i) {
            int j = dir * 3 + i;
            k_resid_ln<<<ROWS / 8, 256, 0, stream>>>(i == 0 ? input : cur, res, hn,
                                                     ln_w + j * DMODEL, ln_b + j * DMODEL,
                                                     i == 0);
            launch_gemm(hn, DMODEL, in_proj_w + (size_t)j * 2 * DINNER * DMODEL, DMODEL,
                        2 * DINNER, xz, 2 * DINNER, DMODEL, stream);
            k_dwconv_silu<<<ROWS * DINNER / 256, 256, 0, stream>>>(
                xz, conv_w + j * DINNER * 4, conv_b + j * DINNER, xcb);
            launch_gemm(xcb, DINNER, xproj_w + (size_t)j * XPN * DINNER, DINNER,
                        XPN, dbc, XPN, DINNER, stream);
            launch_gemm(dbc, XPN, dtproj_w + (size_t)j * DINNER * DTRANK, DTRANK,
                        DINNER, dtf, DINNER, DTRANK, stream);
            k_softplus_bias<<<ROWS * DINNER / 256, 256, 0, stream>>>(dtf, dtproj_b + j * DINNER);
            k_ssm_scan<<<(BATCH * DINNER * NST) / 256, 256, 0, stream>>>(
                dtf, xcb, dbc, xz, A_log + (size_t)j * DINNER * NST, Dparam + j * DINNER, ybuf);
            launch_gemm(ybuf, DINNER, outproj_w + (size_t)j * DMODEL * DINNER, DINNER,
                        DMODEL, cur, DMODEL, DINNER, stream);
        }
        if (dir == 0) k_finish_fwd<<<NH / 256, 256, 0, stream>>>(cur, res, sumf, rf_o);
        else          k_finish_bwd<<<NH / 256, 256, 0, stream>>>(cur, res, sumf, out0, rb_o);
    }
}